// SparseGNBN_25683904430827
// MI455X (gfx1250) — compile-verified
//
#include <hip/hip_runtime.h>

typedef float v2f __attribute__((ext_vector_type(2)));
typedef float v4f __attribute__((ext_vector_type(4)));
typedef float v8f __attribute__((ext_vector_type(8)));

#define EPS 1e-5f
// partial record per block: s1[8][64] | s2[8][64] | cnt[8]  -> 1032 floats
#define PREC 1032

__device__ __forceinline__ v8f wmma_f32(v2f a, v2f b, v8f c) {
    return __builtin_amdgcn_wmma_f32_16x16x4_f32(
        false, a, false, b, (short)0, c, false, false);
}

// Process one K=4 chunk of 4 consecutive points starting at row p (all in-range).
// Branch-free: one b64 batch_idx load + 8 b32 feature loads per lane.
__device__ __forceinline__ void chunk4(const float* __restrict__ feat,
                                       const int* __restrict__ bidx,
                                       long long p, int h, int l,
                                       v8f acc1[4], v8f acc2[4], float& cntAcc) {
    long long r0 = p + 2 * h;                       // this half's first point
    int2 bi = *reinterpret_cast<const int2*>(bidx + r0);   // 8B-aligned (r0 even)
    v2f a;
    a.x = (bi.x == l) ? 1.0f : 0.0f;                // A[m=l][k=2h]
    a.y = (bi.y == l) ? 1.0f : 0.0f;                // A[m=l][k=2h+1]
    cntAcc += a.x + a.y;

    const float* row0 = feat + r0 * 64;
    const float* row1 = row0 + 64;
#pragma unroll
    for (int t = 0; t < 4; ++t) {
        v2f bm;
        bm.x = row0[t * 16 + l];                    // B[k=2h][n=l]
        bm.y = row1[t * 16 + l];                    // B[k=2h+1][n=l]
        v2f bq;
        bq.x = bm.x * bm.x;
        bq.y = bm.y * bm.y;
        acc1[t] = wmma_f32(a, bm, acc1[t]);
        acc2[t] = wmma_f32(a, bq, acc2[t]);
    }
}

// Masked variant for the ragged tail (runs at most twice, on one wave).
// Loads are unconditional from clamped rows; masking lives in the one-hot.
__device__ __forceinline__ void chunk4_masked(const float* __restrict__ feat,
                                              const int* __restrict__ bidx,
                                              long long p, long long N, int h, int l,
                                              v8f acc1[4], v8f acc2[4], float& cntAcc) {
    long long r0 = p + 2 * h;
    long long r1 = r0 + 1;
    long long c0 = (r0 < N) ? r0 : (N - 1);
    long long c1 = (r1 < N) ? r1 : (N - 1);
    int b0 = bidx[c0];
    int b1 = bidx[c1];
    v2f a;
    a.x = ((r0 < N) && (b0 == l)) ? 1.0f : 0.0f;
    a.y = ((r1 < N) && (b1 == l)) ? 1.0f : 0.0f;
    cntAcc += a.x + a.y;

    const float* row0 = feat + c0 * 64;
    const float* row1 = feat + c1 * 64;
#pragma unroll
    for (int t = 0; t < 4; ++t) {
        v2f bm;
        bm.x = row0[t * 16 + l];
        bm.y = row1[t * 16 + l];
        v2f bq;
        bq.x = bm.x * bm.x;
        bq.y = bm.y * bm.y;
        acc1[t] = wmma_f32(a, bm, acc1[t]);   // a==0 kills OOB contributions
        acc2[t] = wmma_f32(a, bq, acc2[t]);
    }
}

// ---------------------------------------------------------------------------
// Stage 1: segment sums via one-hot WMMA  (D = onehot(A) x feat(B) + C)
// Each wave: 8 points (two K=4 chunks) x 64 channels per iteration
//            = 16 v_wmma_f32_16x16x4_f32, branch-free body.
// ---------------------------------------------------------------------------
__global__ void gn_stage1(const float* __restrict__ feat,
                          const int* __restrict__ bidx,
                          int N, float* __restrict__ partials) {
    __shared__ float lds[PREC];   // [0,512) s1[b][c], [512,1024) s2[b][c], [1024,1032) cnt[b]
    for (int i = threadIdx.x; i < PREC; i += blockDim.x) lds[i] = 0.0f;
    __syncthreads();

    const int lane = threadIdx.x & 31;
    const int h    = lane >> 4;        // half: points {2h, 2h+1} of each 4-point chunk
    const int l    = lane & 15;        // A row (batch slot) / B column (channel in tile)
    const int wavesPerBlk = blockDim.x >> 5;
    const int W      = blockIdx.x * wavesPerBlk + (threadIdx.x >> 5);
    const int totalW = gridDim.x * wavesPerBlk;

    v8f acc1[4] = {};
    v8f acc2[4] = {};
    float cntAcc = 0.0f;

    const long long NN = (long long)N;
    const long long stride8 = (long long)totalW * 8;
    long long p = (long long)W * 8;

    for (; p + 8 <= NN; p += stride8) {            // branch-free main loop
        chunk4(feat, bidx, p,     h, l, acc1, acc2, cntAcc);
        chunk4(feat, bidx, p + 4, h, l, acc1, acc2, cntAcc);
    }
    for (; p < NN; p += 4)                          // ragged tail, <=2 iterations
        chunk4_masked(feat, bidx, p, NN, h, l, acc1, acc2, cntAcc);

    // Cross-wave reduction in LDS. Valid rows (M=0..7) live in lanes 0-15.
    if (h == 0) {
#pragma unroll
        for (int t = 0; t < 4; ++t)
#pragma unroll
            for (int r = 0; r < 8; ++r) {
                atomicAdd(&lds[r * 64 + t * 16 + l], acc1[t][r]);
                atomicAdd(&lds[512 + r * 64 + t * 16 + l], acc2[t][r]);
            }
    }
    if (l < 8) atomicAdd(&lds[1024 + l], cntAcc);   // lanes 0-7 and 16-23
    __syncthreads();

    float* out = partials + (size_t)blockIdx.x * PREC;
    for (int i = threadIdx.x; i < PREC; i += blockDim.x) out[i] = lds[i];
}

// ---------------------------------------------------------------------------
// Stage 2: reduce block partials -> mean[256], inv[256]  (index = b*32 + g)
// ---------------------------------------------------------------------------
__global__ void gn_stage2(const float* __restrict__ partials, int numBlocks,
                          float* __restrict__ stats) {
    int t = threadIdx.x;
    if (t >= 256) return;
    int b = t >> 5;
    int g = t & 31;
    float s1 = 0.0f, s2 = 0.0f, cnt = 0.0f;
    for (int k = 0; k < numBlocks; ++k) {
        const float* pp = partials + (size_t)k * PREC;
        s1  += pp[b * 64 + 2 * g] + pp[b * 64 + 2 * g + 1];
        s2  += pp[512 + b * 64 + 2 * g] + pp[512 + b * 64 + 2 * g + 1];
        cnt += pp[1024 + b];
    }
    float c    = fmaxf(cnt * 2.0f, 1.0f);   // Cg = 2, clamp like reference
    float mean = s1 / c;
    float var  = s2 / c - mean * mean;
    float inv  = rsqrtf(var + EPS);
    stats[t]       = mean;
    stats[256 + t] = inv;
}

// ---------------------------------------------------------------------------
// Stage 3: streaming normalize, float4 per thread, stats in LDS,
//          non-temporal stores for the write-once output.
// ---------------------------------------------------------------------------
__global__ void gn_stage3(const float* __restrict__ feat,
                          const int* __restrict__ bidx,
                          const float* __restrict__ weight,
                          const float* __restrict__ bias,
                          const float* __restrict__ stats,
                          float* __restrict__ out, int N) {
    __shared__ float sm[256];
    __shared__ float si[256];
    if (threadIdx.x < 256) {
        sm[threadIdx.x] = stats[threadIdx.x];
        si[threadIdx.x] = stats[256 + threadIdx.x];
    }
    __syncthreads();

    long long total  = (long long)N * 16;   // number of float4 elements
    long long idx    = (long long)blockIdx.x * blockDim.x + threadIdx.x;
    long long stride = (long long)gridDim.x * blockDim.x;   // multiple of 16
    int q = (int)(idx & 15);                // channel-quad, invariant per thread

    v4f w  = ((const v4f*)weight)[q];
    v4f bb = ((const v4f*)bias)[q];
    const v4f* f4 = (const v4f*)feat;
    v4f* o4 = (v4f*)out;

    for (; idx < total; idx += stride) {
        long long r = idx >> 4;
        int bi = bidx[r];
        v4f f = f4[idx];
        int s = bi * 32 + (q << 1);
        float m0 = sm[s], i0 = si[s];
        float m1 = sm[s + 1], i1 = si[s + 1];
        v4f o;
        o.x = (f.x - m0) * i0 * w.x + bb.x;
        o.y = (f.y - m0) * i0 * w.y + bb.y;
        o.z = (f.z - m1) * i1 * w.z + bb.z;
        o.w = (f.w - m1) * i1 * w.w + bb.w;
        __builtin_nontemporal_store(o, &o4[idx]);
    }
}

// ---------------------------------------------------------------------------
extern "C" void kernel_launch(void* const* d_in, const int* in_sizes, int n_in,
                              void* d_out, int out_size, void* d_ws, size_t ws_size,
                              hipStream_t stream) {
    const float* feat   = (const float*)d_in[0];
    const float* weight = (const float*)d_in[1];
    const float* bias   = (const float*)d_in[2];
    const int*   bidx   = (const int*)d_in[3];
    int N = in_sizes[3];
    float* out = (float*)d_out;
    float* ws  = (float*)d_ws;

    int grid1 = 256;
    size_t needFloats = (size_t)grid1 * PREC + 512;
    if (needFloats * sizeof(float) > ws_size) {
        long long avail = (long long)(ws_size / sizeof(float)) - 512;
        long long g = avail / PREC;
        grid1 = (g < 1) ? 1 : (g > 256 ? 256 : (int)g);
    }
    float* partials = ws;
    float* stats    = ws + (size_t)grid1 * PREC;

    gn_stage1<<<grid1, 256, 0, stream>>>(feat, bidx, N, partials);
    gn_stage2<<<1, 256, 0, stream>>>(partials, grid1, stats);

    long long total4 = (long long)N * 16;
    long long g3 = (total4 + 255) / 256;
    int grid3 = (g3 > 8192) ? 8192 : (g3 < 1 ? 1 : (int)g3);
    gn_stage3<<<grid3, 256, 0, stream>>>(feat, bidx, weight, bias, stats, out, N);
}